// NNLoss_41377714929793
// MI455X (gfx1250) — compile-verified
//
#include <hip/hip_runtime.h>
#include <hip/hip_bf16.h>
#include <math.h>

// Chamfer-style bidirectional NN L1 loss for MI455X (gfx1250, wave32).
//
// d2[m][n] = |r_m|^2 + |q_n|^2 - 2 r_m.q_n is produced ENTIRELY inside one
// V_WMMA_F32_16X16X4_F32 per 16x16 tile by packing the rank-1 terms into the
// two otherwise-unused K slots (D=2, K=4):
//    A row m = (-2x_m, -2y_m, |r_m|^2, 1)      (K2,K3 live in lanes 16-31)
//    B col n = ( qx_n,  qy_n, 1, |q_n|^2)
//    C       = 0 (hoisted zero accumulator)
// => D[m][n] = -2 r.q + |r|^2 + |q|^2 = d^2, no C-preload, no extra VALU.
//
// Argmin = depth-3 tournament per lane (8 rows/lane) + one shfl_xor(16)
// lexicographic merge -> exact jnp.argmin first-min tie-breaking.

typedef __attribute__((ext_vector_type(2))) float v2f;
typedef __attribute__((ext_vector_type(4))) float v4f;
typedef __attribute__((ext_vector_type(8))) float v8f;

constexpr int BATCH            = 16;
constexpr int NPTS             = 2048;            // N == M in the reference
constexpr int WAVES            = 8;               // 256-thread block = 8 wave32
constexpr int BLOCKS_PER_BATCH = NPTS / (16 * WAVES); // 16
constexpr int REF_TILES        = NPTS / 16;       // 128
constexpr int PARTIALS_PER_DIR = BATCH * BLOCKS_PER_BATCH * WAVES; // 2048

// Lexicographic (d, idx) merge; i_cur < i_new by construction, so strict '<'
// keeps the lower index on ties (jnp.argmin semantics).
__device__ __forceinline__ void merge_min(float& d, int& i, float dn, int in_) {
    const bool lt = dn < d;
    d = lt ? dn : d;
    i = lt ? in_ : i;
}

__global__ __launch_bounds__(256)
void nn_dir_kernel(const float* __restrict__ qpts,   // [B, NPTS, 2] queries
                   const float* __restrict__ rpts,   // [B, NPTS, 2] refs
                   const float* __restrict__ coef,   // [2] or nullptr (=> 1.0)
                   float* __restrict__ partials)     // [gridDim.x * WAVES]
{
    __shared__ __align__(16) float sOp[NPTS * 4];  // {-2x,-2y,|r|^2,1}  32 KB
    __shared__ __align__(16) float sXY[NPTS * 2];  // refs x,y (gather)  16 KB

    const int b  = blockIdx.x / BLOCKS_PER_BATCH;
    const int bt = blockIdx.x % BLOCKS_PER_BATCH;
    const float* refB = rpts + (size_t)b * NPTS * 2;

    // Stage all refs of this batch as packed WMMA A-operand rows.
    for (int i = threadIdx.x; i < NPTS; i += blockDim.x) {
        const float x = refB[2 * i];
        const float y = refB[2 * i + 1];
        sXY[2 * i]     = x;
        sXY[2 * i + 1] = y;
        v4f op;
        op.x = -2.0f * x;
        op.y = -2.0f * y;
        op.z = x * x + y * y;
        op.w = 1.0f;
        *(v4f*)(sOp + 4 * i) = op;
    }
    __syncthreads();

    const int  lane    = threadIdx.x & 31;
    const int  wave    = threadIdx.x >> 5;
    const int  n       = lane & 15;        // query column / ref row within tile
    const bool loHalf  = lane < 16;
    const int  halfOff = loHalf ? 0 : 8;   // which 8 C rows this half holds

    // This wave's 16 queries; every lane keeps query (lane%16) coords for the
    // final L1 gather. B operand: (qx,qy)=K0,K1 on lo half; (1,|q|^2)=K2,K3 hi.
    const int   qIdx = (bt * WAVES + wave) * 16 + n;
    const float qx   = qpts[((size_t)b * NPTS + qIdx) * 2];
    const float qy   = qpts[((size_t)b * NPTS + qIdx) * 2 + 1];
    const float qSq  = qx * qx + qy * qy;

    v2f bmat;
    bmat.x = loHalf ? qx : 1.0f;
    bmat.y = loHalf ? qy : qSq;

    // A operand: each lane loads one b64 pair of its ref's packed float4:
    // lo lanes -> (-2x,-2y) = K0,K1; hi lanes -> (|r|^2, 1) = K2,K3.
    // Lane n touches banks 4n..4n+3: conflict-free broadcast-free pattern.
    const float* aPtr = sOp + 4 * n + (loHalf ? 0 : 2);

    const v8f czero = {};   // hoisted zero accumulator (C = inline 0)

    float bestD = 3.0e38f;
    int   bestI = 0;

    for (int t = 0; t < REF_TILES; ++t) {
        const v2f amat = *(const v2f*)(aPtr + t * 64);

        // One wmma produces the complete 16x16 d^2 tile (incl. rank-1 terms).
        const v8f c = __builtin_amdgcn_wmma_f32_16x16x4_f32(
                /*neg_a=*/false, amat, /*neg_b=*/false, bmat,
                /*c_mod=*/(short)0, czero, /*reuse_a=*/false, /*reuse_b=*/false);

        // Depth-3 tournament argmin over this lane's 8 ref rows.
        float d0 = c[0]; int i0 = 0; merge_min(d0, i0, c[1], 1);
        float d2 = c[2]; int i2 = 2; merge_min(d2, i2, c[3], 3);
        float d4 = c[4]; int i4 = 4; merge_min(d4, i4, c[5], 5);
        float d6 = c[6]; int i6 = 6; merge_min(d6, i6, c[7], 7);
        merge_min(d0, i0, d2, i2);
        merge_min(d4, i4, d6, i6);
        merge_min(d0, i0, d4, i4);

        // Fold into the running best (earlier tiles have lower indices).
        const int  base = t * 16 + halfOff;
        const bool lt   = d0 < bestD;
        bestD = lt ? d0 : bestD;
        bestI = lt ? (base + i0) : bestI;
    }

    // Merge the two half-wave argmins (disjoint ref-row subsets) with
    // lexicographic (d, idx) order -> exact jnp.argmin tie semantics.
    const float od = __shfl_xor(bestD, 16, 32);
    const int   oi = __shfl_xor(bestI, 16, 32);
    const bool take = (od < bestD) || ((od == bestD) && (oi < bestI));
    bestD = take ? od : bestD;
    bestI = take ? oi : bestI;

    float c0 = 1.0f, c1 = 1.0f;
    if (coef) { c0 = coef[0]; c1 = coef[1]; }

    const float rx = sXY[2 * bestI];
    const float ry = sXY[2 * bestI + 1];
    float contrib  = loHalf ? (fabsf(qx - rx) * c0 + fabsf(qy - ry) * c1) : 0.0f;

    // Fixed-order butterfly sum across the wave (deterministic).
#pragma unroll
    for (int off = 16; off >= 1; off >>= 1)
        contrib += __shfl_xor(contrib, off, 32);

    if (lane == 0)
        partials[blockIdx.x * WAVES + wave] = contrib;
}

__global__ __launch_bounds__(256)
void nn_reduce_kernel(const float* __restrict__ partials, float* __restrict__ out, int n)
{
    __shared__ float s[256];
    float acc = 0.0f;
    for (int i = threadIdx.x; i < n; i += 256) acc += partials[i];
    s[threadIdx.x] = acc;
    __syncthreads();
#pragma unroll
    for (int off = 128; off > 0; off >>= 1) {
        if ((int)threadIdx.x < off) s[threadIdx.x] += s[threadIdx.x + off];
        __syncthreads();
    }
    if (threadIdx.x == 0) out[0] = s[0];
}

extern "C" void kernel_launch(void* const* d_in, const int* in_sizes, int n_in,
                              void* d_out, int out_size, void* d_ws, size_t ws_size,
                              hipStream_t stream) {
    const float* preds   = (const float*)d_in[0];  // [16, 2048, 2] f32
    const float* targs   = (const float*)d_in[1];  // [16, 2048, 2] f32
    const float* subcoef = (const float*)d_in[2];  // [2] f32 (ones)
    float* out      = (float*)d_out;               // scalar f32
    float* partials = (float*)d_ws;                // 2 * 2048 f32 = 16 KB

    dim3 blk(256);
    dim3 grd(BATCH * BLOCKS_PER_BATCH);            // 256 blocks per direction

    // Direction 0: preds -> nearest targ, weighted by subcoef.
    nn_dir_kernel<<<grd, blk, 0, stream>>>(preds, targs, subcoef, partials);
    // Direction 1: targs -> nearest pred, unit weights.
    nn_dir_kernel<<<grd, blk, 0, stream>>>(targs, preds, nullptr,
                                           partials + PARTIALS_PER_DIR);
    // Deterministic fixed-order final reduction.
    nn_reduce_kernel<<<1, 256, 0, stream>>>(partials, out, 2 * PARTIALS_PER_DIR);
}